// PointmapSiLogLoss_71691594105396
// MI455X (gfx1250) — compile-verified
//
#include <hip/hip_runtime.h>
#include <stdint.h>
#include <math.h>

// SiLog pointmap loss, fp32, B=32, C=3, H=W=512. Bandwidth-bound two-pass
// reduction. Uses gfx1250 async global->LDS staging (double-buffered) for the
// streaming passes and a layout-agnostic f32 WMMA for the final reduction.

#define EPSV  (-100.0f)
#define CEPS  (1e-4f)

static constexpr int BATCH   = 32;
static constexpr int CH      = 3;
static constexpr int HW      = 512 * 512;        // 262144 per plane
static constexpr int NBLK    = 64;               // blocks per batch
static constexpr int THREADS = 256;              // 8 waves (wave32)
static constexpr int CHUNK   = HW / NBLK;        // 4096 floats per block
static constexpr int TILE    = THREADS * 4;      // 1024 floats per tile (b128/lane)
static constexpr int NTILES  = CHUNK / TILE;     // 4

// ws layout (32-bit slots):
//  [  0.. 31] enc(min pred)   [ 32.. 63] enc(max pred)
//  [ 64.. 95] enc(min targ)   [ 96..127] enc(max targ)
//  [128..159] sum d (f32)     [160..191] sum d^2 (f32)   [192..223] count (f32)

// ---- monotone float <-> uint encoding for atomic min/max --------------------
__device__ __forceinline__ uint32_t fenc(float f) {
  uint32_t u = __float_as_uint(f);
  return (u & 0x80000000u) ? ~u : (u | 0x80000000u);
}
__device__ __forceinline__ float fdec(uint32_t u) {
  return __uint_as_float((u & 0x80000000u) ? (u & 0x7fffffffu) : ~u);
}

// ---- gfx1250 async global -> LDS (ASYNCcnt-tracked) -------------------------
__device__ __forceinline__ void async_ld_b128(uint32_t lds_off, const float* gp) {
  asm volatile("global_load_async_to_lds_b128 %0, %1, off"
               :: "v"(lds_off), "v"((unsigned long long)(uintptr_t)gp)
               : "memory");
}
__device__ __forceinline__ void wait_async_le2() {
  asm volatile("s_wait_asynccnt 0x2" ::: "memory");
}
__device__ __forceinline__ void wait_async_0() {
  asm volatile("s_wait_asynccnt 0x0" ::: "memory");
}

// ---- kernel 0: (re)initialize workspace + output ----------------------------
__global__ void k_init(uint32_t* __restrict__ ws, float* __restrict__ out) {
  int i = threadIdx.x;
  if (i < BATCH) {
    ws[  0 + i] = 0xFFFFFFFFu;  // min pred identity
    ws[ 32 + i] = 0x00000000u;  // max pred identity
    ws[ 64 + i] = 0xFFFFFFFFu;  // min targ identity
    ws[ 96 + i] = 0x00000000u;  // max targ identity
    ((float*)ws)[128 + i] = 0.0f;
    ((float*)ws)[160 + i] = 0.0f;
    ((float*)ws)[192 + i] = 0.0f;
  }
  if (i == 0) out[0] = 0.0f;
}

// ---- kernel 1: per-batch masked min/max of pred_z and target_z --------------
__global__ void k_minmax(const float* __restrict__ pred,
                         const float* __restrict__ targ,
                         uint32_t* __restrict__ ws) {
  __shared__ float4 bufP[2][THREADS];
  __shared__ float4 bufT[2][THREADS];
  __shared__ float  red[4][8];

  const int b   = blockIdx.y;
  const int blk = blockIdx.x;
  const int tid = threadIdx.x;

  const size_t plane = (size_t)b * (CH * HW) + 2 * (size_t)HW;
  const float* P = pred + plane + (size_t)blk * CHUNK;
  const float* T = targ + plane + (size_t)blk * CHUNK;

  const uint32_t offP[2] = { (uint32_t)(uintptr_t)&bufP[0][tid],
                             (uint32_t)(uintptr_t)&bufP[1][tid] };
  const uint32_t offT[2] = { (uint32_t)(uintptr_t)&bufT[0][tid],
                             (uint32_t)(uintptr_t)&bufT[1][tid] };

  float pmin =  INFINITY, pmax = -INFINITY;
  float tmin =  INFINITY, tmax = -INFINITY;

  // prologue: tile 0 -> buffers[0]
  async_ld_b128(offP[0], P + tid * 4);
  async_ld_b128(offT[0], T + tid * 4);

  for (int t = 0; t < NTILES; ++t) {
    const int cur = t & 1;
    if (t + 1 < NTILES) {
      const int nxt = cur ^ 1;
      async_ld_b128(offP[nxt], P + (t + 1) * TILE + tid * 4);
      async_ld_b128(offT[nxt], T + (t + 1) * TILE + tid * 4);
      wait_async_le2();   // in-order completion: tile t done, tile t+1 in flight
    } else {
      wait_async_0();
    }
    const float4 pv = bufP[cur][tid];
    const float4 tv = bufT[cur][tid];
    if (tv.x > EPSV) { pmin = fminf(pmin, pv.x); pmax = fmaxf(pmax, pv.x);
                       tmin = fminf(tmin, tv.x); tmax = fmaxf(tmax, tv.x); }
    if (tv.y > EPSV) { pmin = fminf(pmin, pv.y); pmax = fmaxf(pmax, pv.y);
                       tmin = fminf(tmin, tv.y); tmax = fmaxf(tmax, tv.y); }
    if (tv.z > EPSV) { pmin = fminf(pmin, pv.z); pmax = fmaxf(pmax, pv.z);
                       tmin = fminf(tmin, tv.z); tmax = fmaxf(tmax, tv.z); }
    if (tv.w > EPSV) { pmin = fminf(pmin, pv.w); pmax = fmaxf(pmax, pv.w);
                       tmin = fminf(tmin, tv.w); tmax = fmaxf(tmax, tv.w); }
  }

  // wave32 reduce
  #pragma unroll
  for (int m = 16; m >= 1; m >>= 1) {
    pmin = fminf(pmin, __shfl_xor(pmin, m, 32));
    pmax = fmaxf(pmax, __shfl_xor(pmax, m, 32));
    tmin = fminf(tmin, __shfl_xor(tmin, m, 32));
    tmax = fmaxf(tmax, __shfl_xor(tmax, m, 32));
  }
  const int lane = tid & 31, wid = tid >> 5;
  if (lane == 0) { red[0][wid] = pmin; red[1][wid] = pmax;
                   red[2][wid] = tmin; red[3][wid] = tmax; }
  __syncthreads();
  if (tid == 0) {
    #pragma unroll
    for (int w = 1; w < THREADS / 32; ++w) {
      red[0][0] = fminf(red[0][0], red[0][w]);
      red[1][0] = fmaxf(red[1][0], red[1][w]);
      red[2][0] = fminf(red[2][0], red[2][w]);
      red[3][0] = fmaxf(red[3][0], red[3][w]);
    }
    // fenc(+inf)/fenc(-inf) are identities vs. the ws init values, so blocks
    // with no valid pixels are harmless.
    atomicMin(&ws[  0 + b], fenc(red[0][0]));
    atomicMax(&ws[ 32 + b], fenc(red[1][0]));
    atomicMin(&ws[ 64 + b], fenc(red[2][0]));
    atomicMax(&ws[ 96 + b], fenc(red[3][0]));
  }
}

// ---- kernel 2: per-batch sums of diff_log, diff_log^2, count ----------------
__global__ void k_sums(const float* __restrict__ pred,
                       const float* __restrict__ targ,
                       uint32_t* __restrict__ ws) {
  __shared__ float4 bufP[2][THREADS];
  __shared__ float4 bufT[2][THREADS];
  __shared__ float  red[3][8];

  const int b   = blockIdx.y;
  const int blk = blockIdx.x;
  const int tid = threadIdx.x;

  const float minP = fdec(ws[  0 + b]);
  const float maxP = fdec(ws[ 32 + b]);
  const float minT = fdec(ws[ 64 + b]);
  const float maxT = fdec(ws[ 96 + b]);
  const float irngP = 1.0f / fmaxf(maxP - minP, 1e-6f);
  const float irngT = 1.0f / fmaxf(maxT - minT, 1e-6f);

  const size_t plane = (size_t)b * (CH * HW) + 2 * (size_t)HW;
  const float* P = pred + plane + (size_t)blk * CHUNK;
  const float* T = targ + plane + (size_t)blk * CHUNK;

  const uint32_t offP[2] = { (uint32_t)(uintptr_t)&bufP[0][tid],
                             (uint32_t)(uintptr_t)&bufP[1][tid] };
  const uint32_t offT[2] = { (uint32_t)(uintptr_t)&bufT[0][tid],
                             (uint32_t)(uintptr_t)&bufT[1][tid] };

  float sd = 0.0f, sd2 = 0.0f, cn = 0.0f;

  async_ld_b128(offP[0], P + tid * 4);
  async_ld_b128(offT[0], T + tid * 4);

  for (int t = 0; t < NTILES; ++t) {
    const int cur = t & 1;
    if (t + 1 < NTILES) {
      const int nxt = cur ^ 1;
      async_ld_b128(offP[nxt], P + (t + 1) * TILE + tid * 4);
      async_ld_b128(offT[nxt], T + (t + 1) * TILE + tid * 4);
      wait_async_le2();
    } else {
      wait_async_0();
    }
    const float4 pv = bufP[cur][tid];
    const float4 tv = bufT[cur][tid];
    const float pa[4] = { pv.x, pv.y, pv.z, pv.w };
    const float ta[4] = { tv.x, tv.y, tv.z, tv.w };
    #pragma unroll
    for (int k = 0; k < 4; ++k) {
      if (ta[k] > EPSV) {                       // vm == valid mask here
        const float pn = (pa[k] - minP) * irngP;
        const float tn = (ta[k] - minT) * irngT;
        const float d  = logf(fmaxf(tn, CEPS)) - logf(fmaxf(pn, CEPS));
        if (!isnan(d)) { sd += d; sd2 = fmaf(d, d, sd2); cn += 1.0f; }
      }
    }
  }

  #pragma unroll
  for (int m = 16; m >= 1; m >>= 1) {
    sd  += __shfl_xor(sd,  m, 32);
    sd2 += __shfl_xor(sd2, m, 32);
    cn  += __shfl_xor(cn,  m, 32);
  }
  const int lane = tid & 31, wid = tid >> 5;
  if (lane == 0) { red[0][wid] = sd; red[1][wid] = sd2; red[2][wid] = cn; }
  __syncthreads();
  if (tid == 0) {
    #pragma unroll
    for (int w = 1; w < THREADS / 32; ++w) {
      red[0][0] += red[0][w]; red[1][0] += red[1][w]; red[2][0] += red[2][w];
    }
    float* wsf = (float*)ws;
    atomicAdd(&wsf[128 + b], red[0][0]);
    atomicAdd(&wsf[160 + b], red[1][0]);
    atomicAdd(&wsf[192 + b], red[2][0]);
  }
}

// ---- kernel 3: per-batch loss + mean (WMMA f32 layout-agnostic reduce) ------
typedef float v2f __attribute__((ext_vector_type(2)));
typedef float v8f __attribute__((ext_vector_type(8)));

__global__ void k_final(const uint32_t* __restrict__ ws, float* __restrict__ out) {
  const float* wsf = (const float*)ws;
  const int b = threadIdx.x;            // 0..31, one full wave, EXEC all 1s

  const float cn  = wsf[192 + b];
  const float m1  = wsf[128 + b] / cn;  // cn==0 -> NaN (matches reference)
  const float m2  = wsf[160 + b] / cn;
  const float lb  = sqrtf(m2 - 0.5f * m1 * m1);

  // D = A(16x4) * ones(4x16) + 0. Sum of all D entries == 16 * sum(A),
  // independent of the A/B VGPR striping, so this reduction is exact in f32
  // without relying on layout details.
  v2f a;   a.x = lb;   a.y = 0.0f;
  v2f one; one.x = 1.0f; one.y = 1.0f;
  v8f c = { 0.0f, 0.0f, 0.0f, 0.0f, 0.0f, 0.0f, 0.0f, 0.0f };
  v8f d = __builtin_amdgcn_wmma_f32_16x16x4_f32(
      /*neg_a=*/false, a, /*neg_b=*/false, one,
      /*c_mod=*/(short)0, c, /*reuse_a=*/false, /*reuse_b=*/false);

  float part = d[0] + d[1] + d[2] + d[3] + d[4] + d[5] + d[6] + d[7];
  #pragma unroll
  for (int m = 16; m >= 1; m >>= 1) part += __shfl_xor(part, m, 32);

  if (b == 0) {
    float loss = part * (1.0f / 16.0f) * (1.0f / (float)BATCH);
    if (isnan(loss))      loss = 0.0f;                               // nan_to_num
    else if (isinf(loss)) loss = copysignf(3.402823466e38f, loss);
    out[0] = loss;
  }
}

// ---- host-side launch -------------------------------------------------------
extern "C" void kernel_launch(void* const* d_in, const int* in_sizes, int n_in,
                              void* d_out, int out_size, void* d_ws, size_t ws_size,
                              hipStream_t stream) {
  const float* pred = (const float*)d_in[0];
  const float* targ = (const float*)d_in[1];
  uint32_t* ws  = (uint32_t*)d_ws;
  float*    out = (float*)d_out;

  k_init<<<1, 64, 0, stream>>>(ws, out);
  dim3 grid(NBLK, BATCH);
  k_minmax<<<grid, THREADS, 0, stream>>>(pred, targ, ws);
  k_sums  <<<grid, THREADS, 0, stream>>>(pred, targ, ws);
  k_final <<<1, 32, 0, stream>>>(ws, out);
}